// MMAKernelLayer_56100862820703
// MI455X (gfx1250) — compile-verified
//
#include <hip/hip_runtime.h>
#include <math.h>

typedef __attribute__((ext_vector_type(2))) float v2f;
typedef __attribute__((ext_vector_type(8))) float v8f;

namespace {
constexpr int kRes = 64;     // grid resolution (64x64 pixels)
constexpr int kNB  = 4;      // batches
constexpr int kNN  = 128;    // boxes per batch
constexpr int kNJ  = 16;     // corner points per box
constexpr float kBWInv = 1.0f / 0.3f;
}

// ---------------------------------------------------------------------------
// Kernel 1: per-(b,n) weight  w2 = (max_{i,j} min_c relu(Ds[j,c]-Bs[i,c]))^2
// 512 independent threads; trivially latency-bound.
// ---------------------------------------------------------------------------
__global__ void __launch_bounds__(256) w2_kernel(const float* __restrict__ Bs,
                                                 const float* __restrict__ Ds,
                                                 float* __restrict__ w2) {
  int idx = blockIdx.x * blockDim.x + threadIdx.x;   // b*128 + n
  if (idx >= kNB * kNN) return;
  const float* bp = Bs + idx * kNJ * 2;
  const float* dp = Ds + idx * kNJ * 2;
  float bx[kNJ], by[kNJ], dx[kNJ], dy[kNJ];
#pragma unroll
  for (int j = 0; j < kNJ; ++j) {
    bx[j] = bp[2 * j + 0];
    by[j] = bp[2 * j + 1];
    dx[j] = dp[2 * j + 0];
    dy[j] = dp[2 * j + 1];
  }
  float w = 0.0f;  // all candidates are relu outputs (>= 0), so 0 is a safe identity
#pragma unroll
  for (int i = 0; i < kNJ; ++i) {
#pragma unroll
    for (int j = 0; j < kNJ; ++j) {
      float r0 = fmaxf(dx[j] - bx[i], 0.0f);
      float r1 = fmaxf(dy[j] - by[i], 0.0f);
      w = fmaxf(w, fminf(r0, r1));
    }
  }
  w2[idx] = w * w;  // P = 2.0; finite inputs -> isfinite guard is identity
}

// ---------------------------------------------------------------------------
// Kernel 2: fused image pass.
//   - One block = 8 waves, all sharing the same batch b (256 tiles/batch,
//     8 | 256). Box data (Bs[b], Ds[b], w2[b]) staged once into LDS.
//   - One wave = one 16-pixel tile. The n-reduction img = k . w^2 is done
//     with V_WMMA_F32_16X16X4_F32: A holds weighted kernel values
//     (M = 16 pixels, K = 4 boxes per chunk, per the documented 32-bit A
//     layout: lanes 0-15 -> K=0,1; lanes 16-31 -> K=2,3), B = all ones so
//     D[i,j] = sum_k A[i,k] independent of B's exact K lane-permutation.
//     C accumulates across the 32 K-chunks (K = 128 total).
// ---------------------------------------------------------------------------
__global__ void __launch_bounds__(256) img_kernel(const float* __restrict__ Bs,
                                                  const float* __restrict__ Ds,
                                                  const float* __restrict__ w2,
                                                  float* __restrict__ out) {
  __shared__ float sB[kNN * kNJ * 2];   // 16 KB
  __shared__ float sD[kNN * kNJ * 2];   // 16 KB
  __shared__ float sW[kNN];             // 512 B

  const int wave   = threadIdx.x >> 5;
  const int lane   = threadIdx.x & 31;
  const int waveId = blockIdx.x * 8 + wave;
  const int b      = waveId >> 8;       // 256 tiles per batch
  const int tile   = waveId & 255;

  // Stage this batch's box data into LDS (float4 coalesced, 4 iters/thread).
  {
    const float4* gB = (const float4*)(Bs + (size_t)b * kNN * kNJ * 2);
    const float4* gD = (const float4*)(Ds + (size_t)b * kNN * kNJ * 2);
    float4* lB = (float4*)sB;
    float4* lD = (float4*)sD;
    for (int i = threadIdx.x; i < kNN * kNJ * 2 / 4; i += 256) {
      lB[i] = gB[i];
      lD[i] = gD[i];
    }
    if (threadIdx.x < kNN) sW[threadIdx.x] = w2[b * kNN + threadIdx.x];
  }
  __syncthreads();

  // Pixel for this lane (WMMA A-matrix M index). Grid is linspace(-3,3,64)
  // with indexing='ij': x0 varies with p/64, x1 with p%64.
  const int li = lane & 15;
  const int hi = lane >> 4;             // which K-pair this lane supplies
  const int p  = tile * 16 + li;
  const float x0 = -3.0f + 6.0f * (float)(p >> 6) / 63.0f;
  const float x1 = -3.0f + 6.0f * (float)(p & 63) / 63.0f;

  v8f acc = {};
  for (int nc = 0; nc < kNN; nc += 4) {
    v2f a;
#pragma unroll
    for (int t = 0; t < 2; ++t) {
      const int n = nc + 2 * hi + t;    // lanes 0-15: K=0,1 ; lanes 16-31: K=2,3
      const float2* Bp = (const float2*)(sB + n * (kNJ * 2));
      const float2* Dp = (const float2*)(sD + n * (kNJ * 2));
      float db = 3.0e38f, dd = 3.0e38f;
#pragma unroll
      for (int j = 0; j < kNJ; ++j) {
        float2 bv = Bp[j];
        float2 dv = Dp[j];
        db = fminf(db, fmaxf(bv.x - x0, bv.y - x1));
        dd = fminf(dd, fmaxf(x0 - dv.x, x1 - dv.y));
      }
      float dist = fmaxf(fmaxf(db, dd), 0.0f) * kBWInv;
      a[t] = expf(-0.5f * dist * dist) * sW[n];   // fold w^2 into A
    }
    v2f ones = {1.0f, 1.0f};                      // B = 1 -> row sums, layout-robust
    // (neg_a, A, neg_b, B, c_mod, C, reuse_a, reuse_b)
    acc = __builtin_amdgcn_wmma_f32_16x16x4_f32(
        false, a, false, ones, (short)0, acc, false, false);
  }

  // C/D layout: VGPR r holds M=r (lanes 0-15) and M=r+8 (lanes 16-31);
  // every column (N) is identical, so lanes 0 and 16 carry the full tile.
  float* op = out + (size_t)b * (kRes * kRes) + tile * 16;
  if (lane == 0) {
#pragma unroll
    for (int r = 0; r < 8; ++r) op[r] = acc[r];
  } else if (lane == 16) {
#pragma unroll
    for (int r = 0; r < 8; ++r) op[8 + r] = acc[r];
  }
}

extern "C" void kernel_launch(void* const* d_in, const int* in_sizes, int n_in,
                              void* d_out, int out_size, void* d_ws, size_t ws_size,
                              hipStream_t stream) {
  (void)in_sizes; (void)n_in; (void)out_size; (void)ws_size;
  const float* Bs = (const float*)d_in[0];   // (4,128,16,2) f32
  const float* Ds = (const float*)d_in[1];   // (4,128,16,2) f32
  float* w2  = (float*)d_ws;                 // 512 floats scratch
  float* out = (float*)d_out;                // (4,4096) f32

  w2_kernel<<<2, 256, 0, stream>>>(Bs, Ds, w2);
  img_kernel<<<128, 256, 0, stream>>>(Bs, Ds, w2, out);
}